// DistanceSelection_9526237463162
// MI455X (gfx1250) — compile-verified
//
#include <hip/hip_runtime.h>

namespace {
constexpr int   kB         = 64;
constexpr int   kN         = 100000;
constexpr int   kF         = 5;
constexpr int   kMaxInc    = 64;
constexpr float kSqCut     = 9.0f;        // CUTOFF^2
constexpr int   kCap       = 256;         // per-batch candidate capacity (E[hits]~11)
constexpr int   kBpb       = 8;           // blocks per batch
constexpr int   kTpb       = 256;         // threads per block (8 waves)
constexpr int   kTileP     = 2048;        // particles per tile
constexpr int   kTileBytes = kTileP * 12;               // 24576 B
constexpr int   kNTiles    = (kN + kTileP - 1) / kTileP; // 49
constexpr int   kBatchB    = kN * 12;                   // 1200000 B (16-divisible)
constexpr int   kChunks    = kTileBytes / (kTpb * 16);  // 6 async b128 per thread per tile
}

// ---- CDNA5 async data-mover path (ASYNCcnt-tracked LDS fill) ----
__device__ __forceinline__ void async_copy_b128(unsigned lds_addr, unsigned voff,
                                                const void* sbase) {
  asm volatile("global_load_async_to_lds_b128 %0, %1, %2"
               :: "v"(lds_addr), "v"(voff), "s"(sbase)
               : "memory");
}
__device__ __forceinline__ void wait_asynccnt_le6() {
  asm volatile("s_wait_asynccnt 0x6" ::: "memory");
}
__device__ __forceinline__ void wait_asynccnt_0() {
  asm volatile("s_wait_asynccnt 0x0" ::: "memory");
}

__global__ void ds_init_kernel(int* __restrict__ counters) {
  if (threadIdx.x < kB) counters[threadIdx.x] = 0;
}

__global__ __launch_bounds__(kTpb)
void ds_scan_kernel(const float* __restrict__ coords,
                    const float* __restrict__ ref,
                    const float* __restrict__ boxl,
                    int*   __restrict__ counters,
                    float* __restrict__ cand_d,
                    int*   __restrict__ cand_i) {
  __shared__ __align__(16) float smem[2][kTileBytes / 4];

  const int b   = blockIdx.x / kBpb;
  const int sub = blockIdx.x % kBpb;
  const int tid = threadIdx.x;

  const char* cbase = (const char*)(coords + (size_t)b * kN * 3);
  const float rx = ref[b * 3 + 0], ry = ref[b * 3 + 1], rz = ref[b * 3 + 2];
  const float bx = boxl[b * 3 + 0], by = boxl[b * 3 + 1], bz = boxl[b * 3 + 2];
  const float ix = 1.0f / bx, iy = 1.0f / by, iz = 1.0f / bz;

  const unsigned lds[2] = { (unsigned)(size_t)&smem[0][0],
                            (unsigned)(size_t)&smem[1][0] };

  auto stage = [&](int tile, int bsel) {
    const int tbase = tile * kTileBytes;
#pragma unroll
    for (int k = 0; k < kChunks; ++k) {
      const int rel = tid * 16 + k * (kTpb * 16);
      int off = tbase + rel;
      off = (off > (kBatchB - 16)) ? (kBatchB - 16) : off;   // aligned tail clamp
      async_copy_b128(lds[bsel] + (unsigned)rel, (unsigned)off, cbase);
    }
  };

  auto consider = [&](int p, float x, float y, float z) {
    if (p < kN) {
      float dx = x - rx; dx = fmaf(-bx, rintf(dx * ix), dx);
      float dy = y - ry; dy = fmaf(-by, rintf(dy * iy), dy);
      float dz = z - rz; dz = fmaf(-bz, rintf(dz * iz), dz);
      const float r2 = fmaf(dx, dx, fmaf(dy, dy, dz * dz));
      if (r2 <= kSqCut) {                      // rare: ~1.1e-4 per particle
        const int pos = atomicAdd(counters + b, 1);
        if (pos < kCap) {
          cand_d[b * kCap + pos] = r2;
          cand_i[b * kCap + pos] = p;
        }
      }
    }
  };

  int t = sub;
  if (t < kNTiles) stage(t, 0);
  int buf = 0;
  for (; t < kNTiles; t += kBpb) {
    const int  tn   = t + kBpb;
    const bool more = (tn < kNTiles);
    if (more) stage(tn, buf ^ 1);            // prefetch next tile into other buffer
    if (more) wait_asynccnt_le6();           // in-order: current tile resident
    else      wait_asynccnt_0();
    __syncthreads();                          // staging visible across waves

    const float* sp = smem[buf];
#pragma unroll
    for (int h = 0; h < kTileP / (kTpb * 4); ++h) {  // 2 groups of 4 particles/thread
      const int g  = tid + h * kTpb;
      const int p0 = t * kTileP + g * 4;
      const float4* q = (const float4*)(sp + g * 12);
      const float4 a = q[0], c = q[1], e = q[2];
      consider(p0 + 0, a.x, a.y, a.z);
      consider(p0 + 1, a.w, c.x, c.y);
      consider(p0 + 2, c.z, c.w, e.x);
      consider(p0 + 3, e.y, e.z, e.w);
    }
    __syncthreads();                          // buffer free before it is restaged
    buf ^= 1;
  }
}

__global__ __launch_bounds__(kTpb)
void ds_select_kernel(const float* __restrict__ coords,
                      const float* __restrict__ ref,
                      const float* __restrict__ boxl,
                      const float* __restrict__ pinfo,
                      const int*   __restrict__ counters,
                      const float* __restrict__ cand_d,
                      const int*   __restrict__ cand_i,
                      float* __restrict__ out) {
  __shared__ float sd[kCap];
  __shared__ int   si[kCap];

  const int b   = blockIdx.x;
  const int tid = threadIdx.x;

  int count = counters[b];
  if (count > kCap) count = kCap;

  float* oc = out + (size_t)b * kMaxInc * 3;
  float* oi = out + (size_t)kB * kMaxInc * 3 + (size_t)b * kMaxInc * kF;

  for (int i = tid; i < kMaxInc * 3;  i += kTpb) oc[i] = 0.0f;
  for (int i = tid; i < kMaxInc * kF; i += kTpb) oi[i] = 0.0f;
  for (int i = tid; i < count; i += kTpb) {
    sd[i] = cand_d[b * kCap + i];
    si[i] = cand_i[b * kCap + i];
  }
  __syncthreads();

  const float rx = ref[b * 3 + 0], ry = ref[b * 3 + 1], rz = ref[b * 3 + 2];
  const float bx = boxl[b * 3 + 0], by = boxl[b * 3 + 1], bz = boxl[b * 3 + 2];
  const float ix = 1.0f / bx, iy = 1.0f / by, iz = 1.0f / bz;

  for (int i = tid; i < count; i += kTpb) {
    const float di = sd[i];
    const int   ii = si[i];
    int rank = 0;
    for (int j = 0; j < count; ++j) {          // top_k order: dist asc, index tie-break
      const float dj = sd[j];
      const int   ij = si[j];
      rank += (dj < di) || ((dj == di) && (ij < ii));
    }
    if (rank < kMaxInc) {
      const float* cp = coords + ((size_t)b * kN + ii) * 3;
      float dx = cp[0] - rx; dx = fmaf(-bx, rintf(dx * ix), dx);
      float dy = cp[1] - ry; dy = fmaf(-by, rintf(dy * iy), dy);
      float dz = cp[2] - rz; dz = fmaf(-bz, rintf(dz * iz), dz);
      oc[rank * 3 + 0] = dx;
      oc[rank * 3 + 1] = dy;
      oc[rank * 3 + 2] = dz;
      const float* pp = pinfo + ((size_t)b * kN + ii) * kF;
#pragma unroll
      for (int f = 0; f < kF; ++f) oi[rank * kF + f] = pp[f];
    }
  }
}

extern "C" void kernel_launch(void* const* d_in, const int* in_sizes, int n_in,
                              void* d_out, int out_size, void* d_ws, size_t ws_size,
                              hipStream_t stream) {
  (void)in_sizes; (void)n_in; (void)out_size; (void)ws_size;
  const float* coords = (const float*)d_in[0];
  const float* ref    = (const float*)d_in[1];
  const float* boxl   = (const float*)d_in[2];
  const float* pinfo  = (const float*)d_in[3];
  float* out = (float*)d_out;

  // workspace: [64 counters][pad to 256B][cand_d f32 64*256][cand_i i32 64*256] ~128KB
  int*   counters = (int*)d_ws;
  float* cand_d   = (float*)((char*)d_ws + 256);
  int*   cand_i   = (int*)((char*)d_ws + 256 + (size_t)kB * kCap * sizeof(float));

  ds_init_kernel<<<1, 64, 0, stream>>>(counters);
  ds_scan_kernel<<<kB * kBpb, kTpb, 0, stream>>>(coords, ref, boxl,
                                                 counters, cand_d, cand_i);
  ds_select_kernel<<<kB, kTpb, 0, stream>>>(coords, ref, boxl, pinfo,
                                            counters, cand_d, cand_i, out);
}